// PRT_26310969655831
// MI455X (gfx1250) — compile-verified
//
#include <hip/hip_runtime.h>
#include <stdint.h>

typedef __attribute__((ext_vector_type(2))) float v2f;
typedef __attribute__((ext_vector_type(8))) float v8f;

#define B_    128
#define S_    256
#define D_    1024
#define SP    254      // S-2 interior positions
#define TS    16       // s-positions (rows) per block tile
#define WAVES 4
#define ROWP  (D_ + 4) // padded LDS row stride (floats) to avoid bank conflicts

__global__ void PRT_zero_kernel(float* out) { out[0] = 0.0f; }

// Async global->LDS copy of one 16-byte packet per lane (CDNA5 ASYNCcnt path).
// vdst VGPR = LDS byte address, vaddr = 64-bit global address, no SADDR.
__device__ __forceinline__ void async_copy_b128(const void* gptr, void* lptr) {
    uint32_t laddr = (uint32_t)(uintptr_t)lptr;   // flat->LDS: offset lives in addr[31:0]
    uint64_t gaddr = (uint64_t)(uintptr_t)gptr;
    asm volatile("global_load_async_to_lds_b128 %0, %1, off"
                 :: "v"(laddr), "v"(gaddr) : "memory");
}

__device__ __forceinline__ void wait_async_zero() {
#if __has_builtin(__builtin_amdgcn_s_wait_asynccnt)
    __builtin_amdgcn_s_wait_asynccnt(0);
#else
    asm volatile("s_wait_asynccnt 0" ::: "memory");
#endif
}

__global__ __launch_bounds__(128)
void PRT_bridge_loss_kernel(const float* __restrict__ bridges,
                            const int*  __restrict__ b_inx,
                            const int*  __restrict__ neg_i,
                            const int*  __restrict__ neg_j,
                            float* __restrict__ out)
{
    extern __shared__ float smem[];
    float* s_pos  = smem;                    // TS * ROWP
    float* s_neg  = smem + TS * ROWP;        // TS * ROWP
    float* s_head = smem + 2 * TS * ROWP;          // D_
    float* s_tail = smem + 2 * TS * ROWP + D_;     // D_
    float* s_part = smem + 2 * TS * ROWP + 2 * D_; // WAVES*2*TS = 128

    const int b    = blockIdx.x;
    const int tile = blockIdx.y;
    const int tid  = threadIdx.x;   // 0..127
    const int wave = tid >> 5;
    const int lane = tid & 31;

    // ------------- stage head / tail rows via async b128 copies --------------
    const float* headg = bridges + (size_t)b * S_ * D_;
    const float* tailg = bridges + ((size_t)b * S_ + (S_ - 1)) * D_;
    for (int i = tid; i < D_ / 4; i += 128) {
        async_copy_b128(headg + 4 * i, s_head + 4 * i);
        async_copy_b128(tailg + 4 * i, s_tail + 4 * i);
    }

    // ------------- stage 16 pos rows + 16 gathered neg rows (async) ----------
    for (int m = 0; m < TS; ++m) {
        int sidx = tile * TS + m;
        int sc   = sidx < SP ? sidx : (SP - 1);          // clamp: keep loads in-bounds
        const float* posg = bridges + ((size_t)b * S_ + (sc + 1)) * D_;
        int ni = neg_i[b * SP + sc];
        int nj = neg_j[b * SP + sc];
        const float* negg = bridges + ((size_t)ni * S_ + nj) * D_;
        float* lp = s_pos + m * ROWP;
        float* ln = s_neg + m * ROWP;
        for (int i = tid; i < D_ / 4; i += 128) {
            async_copy_b128(posg + 4 * i, lp + 4 * i);
            async_copy_b128(negg + 4 * i, ln + 4 * i);
        }
    }
    wait_async_zero();     // this wave's async LDS writes complete
    __syncthreads();       // visible to all waves in the workgroup

    // ---------------- WMMA row-sum reduction ---------------------------------
    // Lane layout for V_WMMA_F32_16X16X4_F32 A-matrix (16x4 f32):
    //   lanes 0-15: M=lane, K={0,1}; lanes 16-31: M=lane-16, K={2,3}
    const int m  = lane & 15;
    const int kk = (lane >> 4) * 2;
    int sidx = tile * TS + m;
    int sc   = sidx < SP ? sidx : (SP - 1);
    const int* binx_row = b_inx + b * S_;
    float th = (float)binx_row[0];
    float tt = (float)binx_row[S_ - 1];
    float tp = (float)binx_row[sc + 1];
    float alpha = (tp - th) / (tt - th);

    v8f cpos = {};
    v8f cneg = {};
    v2f ones; ones.x = 1.0f; ones.y = 1.0f;

    const float* prow = s_pos + m * ROWP;
    const float* nrow = s_neg + m * ROWP;
    const int kbase = wave * (D_ / WAVES);

    for (int kc = 0; kc < D_ / WAVES; kc += 4) {
        int k = kbase + kc + kk;
        float h0 = s_head[k], h1 = s_head[k + 1];
        float t0 = s_tail[k], t1 = s_tail[k + 1];
        float mix0 = h0 + alpha * (t0 - h0);
        float mix1 = h1 + alpha * (t1 - h1);
        float dp0 = prow[k]     - mix0;
        float dp1 = prow[k + 1] - mix1;
        float dn0 = nrow[k]     - mix0;
        float dn1 = nrow[k + 1] - mix1;
        v2f ap; ap.x = dp0 * dp0; ap.y = dp1 * dp1;
        v2f an; an.x = dn0 * dn0; an.y = dn1 * dn1;
        // C[m][n] += sum_k A[m][k]  (B = ones)  -> row sums in every column
        cpos = __builtin_amdgcn_wmma_f32_16x16x4_f32(
            false, ap, false, ones, (short)0, cpos, false, false);
        cneg = __builtin_amdgcn_wmma_f32_16x16x4_f32(
            false, an, false, ones, (short)0, cneg, false, false);
    }

    // D/C layout: lanes 0-15 hold M=0..7 in v[0..7] (col = lane);
    //             lanes 16-31 hold M=8..15.
    if (lane == 0 || lane == 16) {
        int base = (lane >> 4) * 8;
        for (int v = 0; v < 8; ++v) {
            s_part[wave * 32 + base + v]      = cpos[v];
            s_part[wave * 32 + 16 + base + v] = cneg[v];
        }
    }
    __syncthreads();

    // ---------------- combine wave partials + hinge --------------------------
    if (tid < TS) {
        int row   = tid;
        int sidx2 = tile * TS + row;
        if (sidx2 < SP) {
            float sp = 0.0f, sn = 0.0f;
            for (int w = 0; w < WAVES; ++w) {
                sp += s_part[w * 32 + row];
                sn += s_part[w * 32 + 16 + row];
            }
            float thf = (float)binx_row[0];
            float ttf = (float)binx_row[S_ - 1];
            float tpf = (float)binx_row[sidx2 + 1];
            float a   = (tpf - thf) / (ttf - thf);
            float sigma = a * (ttf - tpf);
            float inv = 1.0f / (2.0f * sigma * sigma);
            // pos_dis - neg_dis = (sn - sp) * inv
            float cur = (sn - sp) * inv + 1.0f;  // BETA = 1.0
            float r = cur > 0.0f ? cur : 0.0f;
            atomicAdd(out, r * (1.0f / (float)B_));
        }
    }
}

extern "C" void kernel_launch(void* const* d_in, const int* in_sizes, int n_in,
                              void* d_out, int out_size, void* d_ws, size_t ws_size,
                              hipStream_t stream) {
    const float* bridges = (const float*)d_in[0];
    const int*   b_inx   = (const int*)d_in[1];
    const int*   neg_i   = (const int*)d_in[2];
    const int*   neg_j   = (const int*)d_in[3];
    float* out = (float*)d_out;

    PRT_zero_kernel<<<1, 1, 0, stream>>>(out);

    dim3 grid(B_, (SP + TS - 1) / TS);  // 128 x 16
    size_t lds_bytes = (size_t)(2 * TS * ROWP + 2 * D_ + WAVES * 2 * TS) * sizeof(float);
    PRT_bridge_loss_kernel<<<grid, 128, lds_bytes, stream>>>(bridges, b_inx, neg_i, neg_j, out);
}